// Encoder_60576218742837
// MI455X (gfx1250) — compile-verified
//
#include <hip/hip_runtime.h>

#define DFEAT 128
#define BN_EPS 1e-5f
#define LDSPITCH 132   // 128 + 4 floats pad: row-to-row LDS bank offset of 4 -> conflict-free b64 reads

typedef float v2f __attribute__((ext_vector_type(2)));
typedef float v8f __attribute__((ext_vector_type(8)));

// ---------------- degree / normalization ----------------

__global__ void k_init_deg(float* __restrict__ deg, int n) {
  int i = blockIdx.x * blockDim.x + threadIdx.x;
  if (i < n) deg[i] = 1.0f;  // self-loop contributes 1
}

__global__ void k_deg_accum(const long long* __restrict__ dst, float* deg, int e) {
  int i = blockIdx.x * blockDim.x + threadIdx.x;
  if (i < e) atomicAdd(&deg[dst[i]], 1.0f);
}

__global__ void k_deg_to_dis(float* __restrict__ deg, int n) {
  int i = blockIdx.x * blockDim.x + threadIdx.x;
  if (i < n) deg[i] = rsqrtf(deg[i]);
}

// ---------------- f32 WMMA GEMM: O[n x 128] = A[n x 128] * W[128 x 128] ----------------
// One block = 16 rows x 128 cols. 8 waves; wave w owns the 16x16 tile at col 16*w.
// A tile (16x128) staged once in LDS (8x redundancy removed); W streamed from L2.
// Per ISA 7.12.2 (f32):
//   A 16x4: lanes 0-15 = {K=k, k+1}, lanes 16-31 = {K=k+2, k+3}, M = lane%16
//   B 4x16: VGPR pair mirrors A's K split, N = lane%16
//   C/D 16x16: VGPR r: lanes 0-15 -> M=r, lanes 16-31 -> M=r+8, N = lane%16
__global__ __launch_bounds__(256) void k_gemm_wmma(
    const float* __restrict__ A, const float* __restrict__ W,
    float* __restrict__ O, int nrows) {
  __shared__ float As[16 * LDSPITCH];

  const int wave = threadIdx.x >> 5;
  const int lane = threadIdx.x & 31;
  const int half = lane >> 4;     // 0 or 1
  const int l16  = lane & 15;
  const int row_base = blockIdx.x * 16;
  const int col_base = wave * 16;
  if (row_base >= nrows) return;  // uniform per block -> EXEC stays all-ones

  // Cooperative stage of the 16x128 A tile into LDS (float4 granularity).
  for (int t = threadIdx.x; t < 16 * 32; t += 256) {   // 512 float4s
    const int r  = t >> 5;       // row in tile
    const int cq = t & 31;       // float4 column
    const int gr = (row_base + r < nrows) ? r : 0;     // tail clamp (dup read, overwritten rows unused)
    const float4 v = ((const float4*)(A + (long)(row_base + gr) * DFEAT))[cq];
    ((float4*)(As + r * LDSPITCH))[cq] = v;
  }
  __syncthreads();

  v8f c = {0.f, 0.f, 0.f, 0.f, 0.f, 0.f, 0.f, 0.f};
  const float* __restrict__ wp = W + col_base + l16;
  const float* __restrict__ asrow = As + l16 * LDSPITCH;

  #pragma unroll 4
  for (int k = 0; k < DFEAT; k += 4) {
    const int ka = k + 2 * half;
    v2f a = *(const v2f*)(asrow + ka);        // ds_load_b64, conflict-free
    v2f b;
    b.x = wp[(long)ka * DFEAT];
    b.y = wp[(long)(ka + 1) * DFEAT];
    c = __builtin_amdgcn_wmma_f32_16x16x4_f32(
        /*neg_a=*/false, a, /*neg_b=*/false, b,
        /*c_mod=*/(short)0, c, /*reuse_a=*/false, /*reuse_b=*/false);
  }

  // Store epilogue: straight-line fast path for full tiles (always taken when n%16==0).
  float* __restrict__ op = O + (long)(row_base + 8 * half) * DFEAT + col_base + l16;
  if (row_base + 16 <= nrows) {
    #pragma unroll
    for (int r = 0; r < 8; ++r) op[(long)r * DFEAT] = c[r];
  } else {
    #pragma unroll
    for (int r = 0; r < 8; ++r) {
      if (row_base + 8 * half + r < nrows) op[(long)r * DFEAT] = c[r];
    }
  }
}

// ---------------- agg init: self-loop term + bias ----------------
__global__ void k_self_bias(const float* __restrict__ hw, const float* __restrict__ dis,
                            const float* __restrict__ bias, float* __restrict__ agg,
                            int total) {
  int idx = blockIdx.x * blockDim.x + threadIdx.x;
  if (idx >= total) return;
  const int i = idx >> 7;          // / DFEAT
  const int d = idx & (DFEAT - 1);
  const float s = dis[i];
  agg[idx] = hw[idx] * s * s + bias[d];
}

// ---------------- edge scatter: one wave per edge, float4 per lane ----------------
__global__ void k_scatter(const float* __restrict__ hw,
                          const long long* __restrict__ src,
                          const long long* __restrict__ dst,
                          const float* __restrict__ dis,
                          float* agg, int e) {
  const int eid = blockIdx.x * (blockDim.x >> 5) + (threadIdx.x >> 5);
  const int lane = threadIdx.x & 31;
  if (eid >= e) return;
  const long long s = src[eid];
  const long long t = dst[eid];
  const float w = dis[s] * dis[t];
  const float4 v = ((const float4*)(hw + s * DFEAT))[lane];
  float* ap = agg + t * DFEAT + lane * 4;
  atomicAdd(ap + 0, v.x * w);
  atomicAdd(ap + 1, v.y * w);
  atomicAdd(ap + 2, v.z * w);
  atomicAdd(ap + 3, v.w * w);
}

// ---------------- BatchNorm statistics ----------------
__global__ void k_zero_stats(float* stats) { stats[threadIdx.x] = 0.f; }  // 256 thr

__global__ void k_stats(const float* __restrict__ h, float* stats, int n) {
  const int d = threadIdx.x;  // blockDim == 128
  float s = 0.f, q = 0.f;
  for (int i = blockIdx.x; i < n; i += gridDim.x) {
    const float v = h[(long)i * DFEAT + d];
    s += v;
    q += v * v;
  }
  atomicAdd(&stats[d], s);
  atomicAdd(&stats[DFEAT + d], q);
}

__global__ void k_stats_fin(float* stats, int n) {
  const int d = threadIdx.x;  // 128
  const float inv_n = 1.0f / (float)n;
  const float mu = stats[d] * inv_n;
  float var = stats[DFEAT + d] * inv_n - mu * mu;
  var = fmaxf(var, 0.f);
  stats[2 * DFEAT + d] = mu;
  stats[3 * DFEAT + d] = rsqrtf(var + BN_EPS);
}

__global__ void k_bnrelu(const float* __restrict__ pre, const float* __restrict__ stats,
                         const float* __restrict__ gamma, const float* __restrict__ beta,
                         float* __restrict__ h1, int total) {
  int idx = blockIdx.x * blockDim.x + threadIdx.x;
  if (idx >= total) return;
  const int d = idx & (DFEAT - 1);
  const float v = (pre[idx] - stats[2 * DFEAT + d]) * stats[3 * DFEAT + d] * gamma[d] + beta[d];
  h1[idx] = fmaxf(v, 0.f);
}

__global__ void k_add_inplace(const float* __restrict__ a, float* __restrict__ out, int total) {
  int idx = blockIdx.x * blockDim.x + threadIdx.x;
  if (idx < total) out[idx] += a[idx];
}

// ---------------- launcher ----------------
extern "C" void kernel_launch(void* const* d_in, const int* in_sizes, int n_in,
                              void* d_out, int out_size, void* d_ws, size_t ws_size,
                              hipStream_t stream) {
  const float*     x     = (const float*)d_in[0];
  const long long* ei    = (const long long*)d_in[1];
  const float*     W1    = (const float*)d_in[2];
  const float*     b1    = (const float*)d_in[3];
  const float*     W2    = (const float*)d_in[4];
  const float*     b2    = (const float*)d_in[5];
  const float*     gamma = (const float*)d_in[6];
  const float*     beta  = (const float*)d_in[7];
  float*           out   = (float*)d_out;

  const int n = in_sizes[0] / DFEAT;  // 50000
  const int e = in_sizes[1] / 2;      // 800000
  const long long* srcp = ei;
  const long long* dstp = ei + e;
  const int total = n * DFEAT;

  float* buf0  = (float*)d_ws;                 // hw  [n*D]
  float* buf1  = buf0 + (size_t)n * DFEAT;     // agg [n*D]
  float* dis   = buf1 + (size_t)n * DFEAT;     // [n]
  float* stats = dis + n;                      // [4*D] sum, sumsq, mu, invstd

  const dim3 blk(256);
  const int gb_n    = (n + 255) / 256;
  const int gb_e    = (e + 255) / 256;
  const int gb_nd   = (total + 255) / 256;
  const int gb_edge = (e + 7) / 8;             // 8 edges (waves) per 256-thr block
  const int gb_tile = (n + 15) / 16;

  // symmetric normalization weights
  k_init_deg  <<<gb_n,    blk, 0, stream>>>(dis, n);
  k_deg_accum <<<gb_e,    blk, 0, stream>>>(dstp, dis, e);
  k_deg_to_dis<<<gb_n,    blk, 0, stream>>>(dis, n);
  k_zero_stats<<<1,       256, 0, stream>>>(stats);

  // layer 1: conv -> BN -> ReLU
  k_gemm_wmma <<<gb_tile, 256, 0, stream>>>(x, W1, buf0, n);
  k_self_bias <<<gb_nd,   blk, 0, stream>>>(buf0, dis, b1, buf1, total);
  k_scatter   <<<gb_edge, blk, 0, stream>>>(buf0, srcp, dstp, dis, buf1, e);
  k_stats     <<<512,     128, 0, stream>>>(buf1, stats, n);
  k_stats_fin <<<1,       128, 0, stream>>>(stats, n);
  k_bnrelu    <<<gb_nd,   blk, 0, stream>>>(buf1, stats, gamma, beta, out, total);  // h1 -> d_out

  // layer 2: conv, then skip add (out = h1 + h2)
  k_gemm_wmma <<<gb_tile, 256, 0, stream>>>(out, W2, buf0, n);
  k_self_bias <<<gb_nd,   blk, 0, stream>>>(buf0, dis, b2, buf1, total);
  k_scatter   <<<gb_edge, blk, 0, stream>>>(buf0, srcp, dstp, dis, buf1, e);
  k_add_inplace<<<gb_nd,  blk, 0, stream>>>(buf1, out, total);
}